// UbH_Gconv_18176301597608
// MI455X (gfx1250) — compile-verified
//
#include <hip/hip_runtime.h>

// ---------------------------------------------------------------------------
// Problem constants (from reference)
// ---------------------------------------------------------------------------
#define NN     128
#define CIN    64
#define COUT   64
#define TT     128
#define VV     25
#define TVTV   (TT * VV)            // 3200
#define MTOT   (NN * TVTV)          // 409600 points (n,t,v)
#define INTER  16
#define SS     5
#define LL     3
#define EPSV   1e-5f

typedef __attribute__((ext_vector_type(2))) float v2f;
typedef __attribute__((ext_vector_type(8))) float v8f;

// One fp32 WMMA step: D(16x16) += A(16x4) * B(4x16), fp32 accumulate.
__device__ __forceinline__ v8f wmma_step(v2f a, v2f b, v8f c) {
  return __builtin_amdgcn_wmma_f32_16x16x4_f32(
      /*neg_a=*/false, a, /*neg_b=*/false, b,
      /*c_mod=*/(short)0, c, /*reuse_a=*/false, /*reuse_b=*/false);
}

// 256-thread block reduction of (sum, sumsq). sh must hold 512 floats.
__device__ __forceinline__ void block_reduce2(float& s, float& q, float* sh) {
  const int tid = threadIdx.x;
  sh[tid] = s; sh[256 + tid] = q;
  __syncthreads();
  for (int off = 128; off > 0; off >>= 1) {
    if (tid < off) { sh[tid] += sh[tid + off]; sh[256 + tid] += sh[256 + tid + off]; }
    __syncthreads();
  }
  s = sh[0]; q = sh[256];
}

// ---------------------------------------------------------------------------
// K1: conv_down. xd[o][m] = sum_c W[o][c] * x[n][c][t][v], m=(n,t,v).
// One wave -> one 16(channel) x 16(point) tile via 16 WMMA K-steps.
// ---------------------------------------------------------------------------
__global__ __launch_bounds__(256) void k_down(const float* __restrict__ x,
                                              const float* __restrict__ Wd,
                                              float* __restrict__ xd) {
  __shared__ float Wsh[INTER * CIN];     // 16x64
  const int tid = threadIdx.x;
  for (int i = tid; i < INTER * CIN; i += 256) Wsh[i] = Wd[i];
  __syncthreads();

  const int wave = tid >> 5, lane = tid & 31;
  const int half = lane >> 4, l16 = lane & 15;
  const int m0 = (blockIdx.x * 8 + wave) * 16;
  const int n  = m0 / TVTV;
  const int mm = m0 - n * TVTV;          // offset within (t,v) plane; tile never crosses n
  const float* xb = x + (size_t)n * CIN * TVTV + mm + l16;

  v8f d = {};
  for (int kk = 0; kk < 16; ++kk) {
    const int c0 = kk * 4 + half * 2;
    v2f a = { Wsh[l16 * CIN + c0], Wsh[l16 * CIN + c0 + 1] };
    v2f b = { xb[(size_t)c0 * TVTV], xb[(size_t)(c0 + 1) * TVTV] };
    d = wmma_step(a, b, d);
  }
  float* ob = xd + m0 + l16;
  #pragma unroll
  for (int r = 0; r < 8; ++r) {
    const int ch = r + half * 8;
    ob[(size_t)ch * MTOT] = d[r];
  }
}

// ---------------------------------------------------------------------------
// K2: per-channel BN stats over contiguous [ch][MTOT] data -> (scale, shift).
// One block per channel; fixed-order tree reduce => deterministic.
// ---------------------------------------------------------------------------
__global__ __launch_bounds__(256) void k_stats_contig(const float* __restrict__ data,
                                                      const float* __restrict__ g,
                                                      const float* __restrict__ beta,
                                                      float* __restrict__ ss) {
  __shared__ float red[512];
  const int c = blockIdx.x, tid = threadIdx.x;
  const float* base = data + (size_t)c * MTOT;
  float s = 0.f, q = 0.f;
  for (int i = tid; i < MTOT; i += 256) { float v = base[i]; s += v; q += v * v; }
  block_reduce2(s, q, red);
  if (tid == 0) {
    const float mean = s / (float)MTOT;
    const float var  = q / (float)MTOT - mean * mean;
    const float sc   = g[c] * rsqrtf(var + EPSV);
    ss[2 * c] = sc; ss[2 * c + 1] = beta[c] - mean * sc;
  }
}

// ---------------------------------------------------------------------------
// K3: per-subset 16x16 channel GEMM with BN+ReLU fused into B loads.
// y[s][o][m] = sum_c Wsub[s][o][c] * relu(BN(xd[c][m])).
// ---------------------------------------------------------------------------
__global__ __launch_bounds__(256) void k_sub(const float* __restrict__ xd,
                                             const float* __restrict__ Wsub,
                                             const float* __restrict__ ssd,
                                             float* __restrict__ y) {
  __shared__ float Wsh[INTER * INTER];
  __shared__ float ss[2 * INTER];
  const int s = blockIdx.y;
  const int tid = threadIdx.x;
  if (tid < INTER * INTER) Wsh[tid] = Wsub[s * INTER * INTER + tid];
  if (tid < 2 * INTER) ss[tid] = ssd[tid];
  __syncthreads();

  const int wave = tid >> 5, lane = tid & 31;
  const int half = lane >> 4, l16 = lane & 15;
  const int m0 = (blockIdx.x * 8 + wave) * 16;
  const float* xb = xd + m0 + l16;

  v8f d = {};
  for (int kk = 0; kk < 4; ++kk) {
    const int c0 = kk * 4 + half * 2;
    v2f a = { Wsh[l16 * INTER + c0], Wsh[l16 * INTER + c0 + 1] };
    const float b0 = fmaxf(xb[(size_t)c0 * MTOT] * ss[2 * c0] + ss[2 * c0 + 1], 0.f);
    const float b1 = fmaxf(xb[(size_t)(c0 + 1) * MTOT] * ss[2 * (c0 + 1)] + ss[2 * (c0 + 1) + 1], 0.f);
    v2f b = { b0, b1 };
    d = wmma_step(a, b, d);
  }
  float* ob = y + ((size_t)s * INTER) * MTOT + m0 + l16;
  #pragma unroll
  for (int r = 0; r < 8; ++r) {
    const int ch = r + half * 8;
    ob[(size_t)ch * MTOT] = d[r];
  }
}

// ---------------------------------------------------------------------------
// K4: adjacency GEMM for stats only. z[t][v] = sum_u y[s][o][n,t,u]*A[s][v][u].
// Block = (s,n,o), 8 waves = all t-tiles. Accumulate per-(s,o,n) sum/sumsq.
// ---------------------------------------------------------------------------
__global__ __launch_bounds__(256) void k_zstat(const float* __restrict__ y,
                                               const float* __restrict__ PA,
                                               float* __restrict__ part) {
  __shared__ float Ash[VV * VV];
  __shared__ float red[512];
  const int bx = blockIdx.x;                 // 5 * 128 * 16
  const int s = bx / (NN * INTER);
  const int rem = bx % (NN * INTER);
  const int n = rem / INTER, o = rem % INTER;
  const int tid = threadIdx.x;
  for (int i = tid; i < VV * VV; i += 256) Ash[i] = PA[s * VV * VV + i];
  __syncthreads();

  const int wave = tid >> 5, lane = tid & 31;
  const int half = lane >> 4, l16 = lane & 15;
  const int t0 = wave * 16;
  const float* yb = y + ((size_t)(s * INTER + o)) * MTOT + (size_t)n * TVTV;

  float lsum = 0.f, lsq = 0.f;
  for (int vt = 0; vt < 2; ++vt) {
    const int v = vt * 16 + l16;
    const bool vc = (v < VV);
    v8f d = {};
    for (int kk = 0; kk < 7; ++kk) {               // K = 25 padded to 28
      const int u0 = kk * 4 + half * 2;
      const bool i0 = (u0 < VV), i1 = (u0 + 1 < VV);
      const int trow = t0 + l16;
      v2f a = { i0 ? yb[trow * VV + u0] : 0.f, i1 ? yb[trow * VV + u0 + 1] : 0.f };
      v2f b = { (i0 && vc) ? Ash[v * VV + u0] : 0.f,
                (i1 && vc) ? Ash[v * VV + u0 + 1] : 0.f };
      d = wmma_step(a, b, d);
    }
    if (vc) {
      #pragma unroll
      for (int r = 0; r < 8; ++r) { lsum += d[r]; lsq += d[r] * d[r]; }
    }
  }
  block_reduce2(lsum, lsq, red);
  if (tid == 0) {
    const int p = (s * INTER + o) * NN + n;
    part[2 * p] = lsum; part[2 * p + 1] = lsq;
  }
}

// K4b: reduce 128 partials per (s,o) -> (scale, shift). 80 blocks, 128 threads.
__global__ __launch_bounds__(128) void k_zreduce(const float* __restrict__ part,
                                                 const float* __restrict__ g,
                                                 const float* __restrict__ beta,
                                                 float* __restrict__ ss) {
  __shared__ float red[256];
  const int so = blockIdx.x, tid = threadIdx.x;
  float s = part[2 * (so * NN + tid)];
  float q = part[2 * (so * NN + tid) + 1];
  red[tid] = s; red[128 + tid] = q;
  __syncthreads();
  for (int off = 64; off > 0; off >>= 1) {
    if (tid < off) { red[tid] += red[tid + off]; red[128 + tid] += red[128 + tid + off]; }
    __syncthreads();
  }
  if (tid == 0) {
    const float mean = red[0] / (float)MTOT;
    const float var  = red[128] / (float)MTOT - mean * mean;
    const float sc   = g[so] * rsqrtf(var + EPSV);
    ss[2 * so] = sc; ss[2 * so + 1] = beta[so] - mean * sc;
  }
}

// ---------------------------------------------------------------------------
// K5: aggregation. For s in 1..4: acc[n][(s-1)*16+o][t][v] += BN(z_s) + BN(z_0).
// Each block recomputes both z_s and z_0 tiles -> unique writer, no races.
// ---------------------------------------------------------------------------
__global__ __launch_bounds__(256) void k_agg(const float* __restrict__ y,
                                             const float* __restrict__ PA,
                                             const float* __restrict__ ssz,
                                             float* __restrict__ acc) {
  __shared__ float As[VV * VV];
  __shared__ float A0[VV * VV];
  const int bx = blockIdx.x;                 // 4 * 128 * 16
  const int si = bx / (NN * INTER);          // 0..3
  const int s  = si + 1;
  const int rem = bx % (NN * INTER);
  const int n = rem / INTER, o = rem % INTER;
  const int tid = threadIdx.x;
  for (int i = tid; i < VV * VV; i += 256) { As[i] = PA[s * VV * VV + i]; A0[i] = PA[i]; }
  __syncthreads();

  const float sc_s = ssz[2 * (s * INTER + o)], sh_s = ssz[2 * (s * INTER + o) + 1];
  const float sc_0 = ssz[2 * o],              sh_0 = ssz[2 * o + 1];

  const int wave = tid >> 5, lane = tid & 31;
  const int half = lane >> 4, l16 = lane & 15;
  const int t0 = wave * 16;
  const float* ybs = y + ((size_t)(s * INTER + o)) * MTOT + (size_t)n * TVTV;
  const float* yb0 = y + ((size_t)o) * MTOT + (size_t)n * TVTV;
  const int c = si * INTER + o;
  float* ab = acc + (size_t)n * COUT * TVTV + (size_t)c * TVTV;

  for (int vt = 0; vt < 2; ++vt) {
    const int v = vt * 16 + l16;
    const bool vc = (v < VV);
    v8f ds = {}, d0 = {};
    for (int kk = 0; kk < 7; ++kk) {
      const int u0 = kk * 4 + half * 2;
      const bool i0 = (u0 < VV), i1 = (u0 + 1 < VV);
      const int trow = t0 + l16;
      v2f a_s = { i0 ? ybs[trow * VV + u0] : 0.f, i1 ? ybs[trow * VV + u0 + 1] : 0.f };
      v2f a_0 = { i0 ? yb0[trow * VV + u0] : 0.f, i1 ? yb0[trow * VV + u0 + 1] : 0.f };
      v2f b_s = { (i0 && vc) ? As[v * VV + u0] : 0.f, (i1 && vc) ? As[v * VV + u0 + 1] : 0.f };
      v2f b_0 = { (i0 && vc) ? A0[v * VV + u0] : 0.f, (i1 && vc) ? A0[v * VV + u0 + 1] : 0.f };
      ds = wmma_step(a_s, b_s, ds);
      d0 = wmma_step(a_0, b_0, d0);
    }
    if (vc) {
      #pragma unroll
      for (int r = 0; r < 8; ++r) {
        const int t = t0 + r + half * 8;
        const float val = (ds[r] * sc_s + sh_s) + (d0[r] * sc_0 + sh_0);
        ab[t * VV + v] += val;
      }
    }
  }
}

// ---------------------------------------------------------------------------
// K6: final BN stats over acc (layout [n][c][t][v]); one block per channel.
// ---------------------------------------------------------------------------
__global__ __launch_bounds__(256) void k_ostat(const float* __restrict__ acc,
                                               const float* __restrict__ g,
                                               const float* __restrict__ beta,
                                               float* __restrict__ ss) {
  __shared__ float red[512];
  const int c = blockIdx.x, tid = threadIdx.x;
  float s = 0.f, q = 0.f;
  for (int i = tid; i < NN * TVTV; i += 256) {
    const int n = i / TVTV, j = i - n * TVTV;
    const float v = acc[(size_t)n * COUT * TVTV + (size_t)c * TVTV + j];
    s += v; q += v * v;
  }
  block_reduce2(s, q, red);
  if (tid == 0) {
    const float mean = s / (float)MTOT;
    const float var  = q / (float)MTOT - mean * mean;
    const float sc   = g[c] * rsqrtf(var + EPSV);
    ss[2 * c] = sc; ss[2 * c + 1] = beta[c] - mean * sc;
  }
}

// K7: out = relu(BN(acc) + x), fused elementwise.
__global__ __launch_bounds__(256) void k_final(const float* __restrict__ acc,
                                               const float* __restrict__ x,
                                               const float* __restrict__ ss,
                                               float* __restrict__ out) {
  const size_t idx = (size_t)blockIdx.x * 256 + threadIdx.x;
  if (idx >= (size_t)NN * COUT * TVTV) return;
  const int c = (int)((idx / TVTV) & (COUT - 1));
  const float v = acc[idx] * ss[2 * c] + ss[2 * c + 1] + x[idx];
  out[idx] = fmaxf(v, 0.f);
}

// ---------------------------------------------------------------------------
// Launch
// ---------------------------------------------------------------------------
extern "C" void kernel_launch(void* const* d_in, const int* in_sizes, int n_in,
                              void* d_out, int out_size, void* d_ws, size_t ws_size,
                              hipStream_t stream) {
  const float* x         = (const float*)d_in[0];
  const float* PA        = (const float*)d_in[1];
  const float* W_down    = (const float*)d_in[2];
  // d_in[3] = b_down: cancels exactly under mean-subtracting BN -> unused
  const float* g_down    = (const float*)d_in[4];
  const float* beta_down = (const float*)d_in[5];
  const float* W_sub     = (const float*)d_in[6];
  // d_in[7] = b_sub: cancels under BN -> unused
  const float* g_sub     = (const float*)d_in[8];
  const float* beta_sub  = (const float*)d_in[9];
  const float* g_out     = (const float*)d_in[10];
  const float* beta_out  = (const float*)d_in[11];
  float* out = (float*)d_out;

  // Workspace layout (floats)
  float* ws    = (float*)d_ws;
  float* xd    = ws;                                  // 16 * MTOT   =  6,553,600
  float* y     = xd + (size_t)INTER * MTOT;           // 80 * MTOT   = 32,768,000
  float* acc   = y + (size_t)SS * INTER * MTOT;       // 64 * MTOT   = 26,214,400
  float* ssd   = acc + (size_t)COUT * MTOT;           // 32
  float* zpart = ssd + 32;                            // 5*16*128*2  = 20,480
  float* ssz   = zpart + 20480;                       // 160
  float* sso   = ssz + 160;                           // 128

  hipMemsetAsync(acc, 0, (size_t)COUT * MTOT * sizeof(float), stream);

  for (int l = 0; l < LL; ++l) {
    k_down<<<MTOT / 16 / 8, 256, 0, stream>>>(x, W_down + l * INTER * CIN, xd);
    k_stats_contig<<<INTER, 256, 0, stream>>>(xd, g_down + l * INTER,
                                              beta_down + l * INTER, ssd);
    dim3 gs(MTOT / 16 / 8, SS);
    k_sub<<<gs, 256, 0, stream>>>(xd, W_sub + l * SS * INTER * INTER, ssd, y);
    k_zstat<<<SS * NN * INTER, 256, 0, stream>>>(y, PA + l * SS * VV * VV, zpart);
    k_zreduce<<<SS * INTER, 128, 0, stream>>>(zpart, g_sub + l * SS * INTER,
                                              beta_sub + l * SS * INTER, ssz);
    k_agg<<<(SS - 1) * NN * INTER, 256, 0, stream>>>(y, PA + l * SS * VV * VV, ssz, acc);
  }
  k_ostat<<<COUT, 256, 0, stream>>>(acc, g_out, beta_out, sso);
  const size_t tot = (size_t)NN * COUT * TVTV;
  k_final<<<(unsigned)((tot + 255) / 256), 256, 0, stream>>>(acc, x, sso, out);
}